// Shift_62577673503005
// MI455X (gfx1250) — compile-verified
//
#include <hip/hip_runtime.h>
#include <hip/hip_bf16.h>

typedef float v2f __attribute__((ext_vector_type(2)));
typedef float v8f __attribute__((ext_vector_type(8)));

#define NB   16
#define TT   1024
#define DD   2000
#define KK   21
#define PP   10
#define RR   (NB * TT)      // 16384 rows
#define WPAD 32             // padded w row stride (floats)

// ---------------------------------------------------------------------------
// Pad W (2000 x 21) into Wp (2000 x 32), zero-filling cols 21..31, so the GEMM
// kernel's B-fragment loads need no bounds guard (keeps EXEC all-ones around
// the WMMAs, no saveexec in the hot loop).
// ---------------------------------------------------------------------------
__global__ __launch_bounds__(32) void pad_w_kernel(
    const float* __restrict__ W, float* __restrict__ Wp) {
  const int k = blockIdx.x;           // 0..1999
  const int c = threadIdx.x;          // 0..31
  Wp[(long)k * WPAD + c] = (c < KK) ? W[(long)k * KK + c] : 0.f;
}

// ---------------------------------------------------------------------------
// GEMM phase:  w[R x 21] = A[R x 2000] @ Wp[2000 x 32]  (V_WMMA_F32_16X16X4_F32)
// A column k comes from Alo for k < split, else Ahi (both row stride DD);
// the loop is split at `split` so the pointer choice is hoisted entirely.
// 4 waves/block, one 16-row tile per wave, two N-tiles (cols 0..15 / 16..31).
// ---------------------------------------------------------------------------
__device__ __forceinline__ void gemm_span(
    const float* __restrict__ A, const float* __restrict__ Wp,
    long rowA, int l15, int half, int kBeg, int kEnd,
    v8f& acc0, v8f& acc1) {
#pragma unroll 2
  for (int k0 = kBeg; k0 < kEnd; k0 += 4) {
    const int kk = k0 + half * 2;
    // A fragment: 16x4 f32; lanes 0-15 hold K=k0,k0+1; lanes 16-31 K=k0+2,k0+3
    v2f a;
    a.x = A[rowA * DD + kk];
    a.y = A[rowA * DD + kk + 1];
    // B fragments from padded W: rows kk,kk+1 (lanes 0-15) / kk+2,kk+3 (16-31)
    v2f b0, b1;
    b0.x = Wp[(long)kk * WPAD + l15];
    b0.y = Wp[(long)(kk + 1) * WPAD + l15];
    b1.x = Wp[(long)kk * WPAD + 16 + l15];
    b1.y = Wp[(long)(kk + 1) * WPAD + 16 + l15];

    acc0 = __builtin_amdgcn_wmma_f32_16x16x4_f32(
        false, a, false, b0, (short)0, acc0, false, false);
    acc1 = __builtin_amdgcn_wmma_f32_16x16x4_f32(
        false, a, false, b1, (short)0, acc1, false, false);
  }
}

__global__ __launch_bounds__(128) void wmma_xw_kernel(
    const float* __restrict__ Alo, const float* __restrict__ Ahi, int split,
    const float* __restrict__ Wp, float* __restrict__ wout) {
  const int lane = threadIdx.x & 31;
  const int wave = threadIdx.x >> 5;
  const int tile = blockIdx.x * 4 + wave;     // 0 .. R/16-1
  const int half = lane >> 4;
  const int l15  = lane & 15;
  const long rowA = (long)(tile * 16 + l15);

  v8f acc0 = {0.f, 0.f, 0.f, 0.f, 0.f, 0.f, 0.f, 0.f};
  v8f acc1 = {0.f, 0.f, 0.f, 0.f, 0.f, 0.f, 0.f, 0.f};

  gemm_span(Alo, Wp, rowA, l15, half, 0,     split, acc0, acc1);
  gemm_span(Ahi, Wp, rowA, l15, half, split, DD,    acc0, acc1);

  // D layout: VGPR v, lanes 0-15 -> M=v ; lanes 16-31 -> M=v+8
#pragma unroll
  for (int v = 0; v < 8; ++v) {
    const long rowO = (long)(tile * 16 + v + half * 8);
    wout[rowO * WPAD + l15]      = acc0[v];
    wout[rowO * WPAD + 16 + l15] = acc1[v];
  }
}

// ---------------------------------------------------------------------------
// Shift-blend phase: out[idx, sOff+d] = sum_k w[idx,k] * x[n, t+k-P, sOff+d]
// One block per (n,t) row, weights in LDS, float4-vectorized streaming
// (all row bases are 16B-aligned: DD, sOff multiples of 4).
// ---------------------------------------------------------------------------
__global__ __launch_bounds__(256) void shift_kernel(
    const float* __restrict__ x, const float* __restrict__ w,
    float* __restrict__ out, int sOff) {
  const int idx  = blockIdx.x;          // 0 .. R-1  (n*T + t)
  const int nIdx = idx / TT;
  const int t    = idx - nIdx * TT;

  __shared__ float sw[KK];
  if (threadIdx.x < KK) sw[threadIdx.x] = w[(long)idx * WPAD + threadIdx.x];
  __syncthreads();

  // valid k range: 0 <= t + k - P < T
  const int kLo   = (t < PP) ? (PP - t) : 0;
  const int kHiEx = (t + PP >= TT) ? (TT - t + PP) : KK;

  for (int d4 = threadIdx.x; d4 < 250; d4 += 256) {
    float4 acc = make_float4(0.f, 0.f, 0.f, 0.f);
    for (int k = kLo; k < kHiEx; ++k) {
      const int ts = t + k - PP;
      const float4 v = ((const float4*)(x + ((long)nIdx * TT + ts) * DD + sOff))[d4];
      const float  s = sw[k];
      acc.x += s * v.x; acc.y += s * v.y; acc.z += s * v.z; acc.w += s * v.w;
    }
    ((float4*)(out + (long)idx * DD + sOff))[d4] = acc;
  }
}

// ---------------------------------------------------------------------------
extern "C" void kernel_launch(void* const* d_in, const int* in_sizes, int n_in,
                              void* d_out, int out_size, void* d_ws, size_t ws_size,
                              hipStream_t stream) {
  const float* x  = (const float*)d_in[0];   // (16,1024,2000)
  const float* W0 = (const float*)d_in[1];   // (2000,21)
  const float* W1 = (const float*)d_in[2];   // (2000,21)
  float* out  = (float*)d_out;               // (16,1024,2000)

  float* wbuf = (float*)d_ws;                          // R  x 32 floats (2 MB)
  float* Wp   = (float*)d_ws + (long)RR * WPAD;        // DD x 32 floats (256 KB)

  (void)in_sizes; (void)n_in; (void)out_size; (void)ws_size;

  // Group 0: w0 = x @ W0 ; out[:, :, 0:1000] = shift(w0, x[:, :, 0:1000])
  pad_w_kernel<<<DD, 32, 0, stream>>>(W0, Wp);
  wmma_xw_kernel<<<RR / 16 / 4, 128, 0, stream>>>(x, x, DD, Wp, wbuf);
  shift_kernel<<<RR, 256, 0, stream>>>(x, wbuf, out, 0);

  // Group 1: w1 = [out0 | x_hi] @ W1 ; out[:, :, 1000:2000] = shift(w1, x_hi)
  pad_w_kernel<<<DD, 32, 0, stream>>>(W1, Wp);
  wmma_xw_kernel<<<RR / 16 / 4, 128, 0, stream>>>(out, x, 1000, Wp, wbuf);
  shift_kernel<<<RR, 256, 0, stream>>>(x, wbuf, out, 1000);
}